// Baseline_mb_69569880261334
// MI455X (gfx1250) — compile-verified
//
#include <hip/hip_runtime.h>
#include <math.h>

typedef __attribute__((ext_vector_type(2))) float v2f;
typedef __attribute__((ext_vector_type(8))) float v8f;

#define HDIM 64
#define G3   192   /* 3*H */
#define PLEN 8
#define DEG  80
#define GITERS 4

__device__ __forceinline__ float gelu_f(float x) {
  return 0.5f * x * (1.0f + erff(x * 0.70710678118654752f));
}
__device__ __forceinline__ float sigmoid_f(float x) {
  return 1.0f / (1.0f + __expf(-x));
}
__device__ __forceinline__ float softplus_f(float x) {
  return fmaxf(x, 0.0f) + log1pf(__expf(-fabsf(x)));
}

// ---------------------------------------------------------------------------
// One 16x16 gate tile pair: ax = X@K[:,c-tile]+b0, ah = H@RK[:,c-tile]+b1
// using V_WMMA_F32_16X16X4_F32 chained over 16 K-slices.
// X A-fragments come from registers; H A-fragments are re-read from the
// per-wave LDS h-tile each slice (keeps VGPR pressure low, avoids spills).
// ---------------------------------------------------------------------------
__device__ __forceinline__ void gate_tile(int c,
                                          const v2f* __restrict__ xfrag,
                                          const float* __restrict__ ht,
                                          const float* __restrict__ KT,
                                          const float* __restrict__ RKT,
                                          const float* __restrict__ b0,
                                          const float* __restrict__ b1,
                                          int coln, int half, int afo,
                                          v8f& axo, v8f& aho)
{
  const float bx = b0[c * 16 + coln];
  const float bh = b1[c * 16 + coln];
  v8f ax = {bx, bx, bx, bx, bx, bx, bx, bx};
  v8f ah = {bh, bh, bh, bh, bh, bh, bh, bh};
  const float* kt  = KT  + (c * 16 + coln) * HDIM + half * 2;
  const float* rkt = RKT + (c * 16 + coln) * HDIM + half * 2;
#pragma unroll
  for (int kk = 0; kk < 16; ++kk) {
    v2f hfr = *(const v2f*)(ht + afo + 4 * kk);
    v2f wk  = *(const v2f*)(kt  + 4 * kk);
    v2f wr  = *(const v2f*)(rkt + 4 * kk);
    ax = __builtin_amdgcn_wmma_f32_16x16x4_f32(false, xfrag[kk], false, wk,
                                               (short)0, ax, false, false);
    ah = __builtin_amdgcn_wmma_f32_16x16x4_f32(false, hfr, false, wr,
                                               (short)0, ah, false, false);
  }
  axo = ax;
  aho = ah;
}

// ---------------------------------------------------------------------------
// Wave-level GRU step on a 16-row tile (H = 64).
//   xb : per-lane pointer to X A-fragment base: lane L holds row (L&15),
//        elements half*2 + {0,1} of each 4-wide K slice (frag kk at +4*kk).
//   ht : per-wave LDS tile, 16x64 row-major, holds h; updated in place.
//   KT/RKT : LDS transposed weights [192][64]  (KT[col][k] = K[k][col]).
//   b0/b1  : LDS biases [192] each (xg bias, hg bias).
// Phase A: r gate + candidate per column group (r consumed immediately).
// Phase B: all z gates (old h still intact in LDS).
// Phase C: h_new = z*h + (1-z)*hh, written back to LDS.
// ---------------------------------------------------------------------------
__device__ __forceinline__ void gru_step_wave(const float* __restrict__ xb,
                                              float* __restrict__ ht,
                                              const float* __restrict__ KT,
                                              const float* __restrict__ RKT,
                                              const float* __restrict__ b0,
                                              const float* __restrict__ b1,
                                              int lane)
{
  const int half = (lane >> 4) & 1;
  const int coln = lane & 15;
  const int afo  = coln * HDIM + half * 2;

  v2f xfrag[16];
#pragma unroll
  for (int kk = 0; kk < 16; ++kk) xfrag[kk] = *(const v2f*)(xb + 4 * kk);

  v8f hh[4];
  v8f zg[4];

  // Phase A: r gates (cols 64..127) and candidates (cols 128..191)
#pragma unroll
  for (int cc = 0; cc < 4; ++cc) {
    v8f ax, ah;
    gate_tile(4 + cc, xfrag, ht, KT, RKT, b0, b1, coln, half, afo, ax, ah);
    v8f r;
#pragma unroll
    for (int v = 0; v < 8; ++v) r[v] = sigmoid_f(ax[v] + ah[v]);
    gate_tile(8 + cc, xfrag, ht, KT, RKT, b0, b1, coln, half, afo, ax, ah);
#pragma unroll
    for (int v = 0; v < 8; ++v) hh[cc][v] = tanhf(ax[v] + r[v] * ah[v]);
  }

  // Phase B: z gates (cols 0..63) -- must finish before h is overwritten
#pragma unroll
  for (int cc = 0; cc < 4; ++cc) {
    v8f ax, ah;
    gate_tile(cc, xfrag, ht, KT, RKT, b0, b1, coln, half, afo, ax, ah);
#pragma unroll
    for (int v = 0; v < 8; ++v) zg[cc][v] = sigmoid_f(ax[v] + ah[v]);
  }

  // Phase C: state update in C-layout
#pragma unroll
  for (int cc = 0; cc < 4; ++cc) {
#pragma unroll
    for (int v = 0; v < 8; ++v) {
      const int   m    = v + half * 8;
      const float hold = ht[m * HDIM + cc * 16 + coln];
      ht[m * HDIM + cc * 16 + coln] =
          zg[cc][v] * hold + (1.0f - zg[cc][v]) * hh[cc][v];
    }
  }
}

#define SMEM_FLOATS (2 * HDIM * G3 + 2 * G3 + 16 * 16 * HDIM)

// ---------------------------------------------------------------------------
// Path RNN: one wave = 16 paths; 8 GRU steps; writes path_state_sequence.
// 165 KB LDS => one block per WGP; allow full VGPR budget.
// ---------------------------------------------------------------------------
__global__ void __launch_bounds__(512, 1)
path_rnn_kernel(const float* __restrict__ link_state,
                const int*   __restrict__ l2p,
                float* __restrict__ path_state,
                float* __restrict__ seq,
                const float* __restrict__ puK,
                const float* __restrict__ puRK,
                const float* __restrict__ pub,
                int n_tiles)
{
  extern __shared__ float smem[];
  float* KT  = smem;
  float* RKT = KT + HDIM * G3;
  float* b0  = RKT + HDIM * G3;
  float* b1  = b0 + G3;
  float* htiles = b1 + G3;

  const int tid = threadIdx.x;
  for (int i = tid; i < HDIM * G3; i += blockDim.x) {
    int r = i / G3, c = i % G3;
    KT[c * HDIM + r]  = puK[i];
    RKT[c * HDIM + r] = puRK[i];
  }
  for (int i = tid; i < G3; i += blockDim.x) { b0[i] = pub[i]; b1[i] = pub[G3 + i]; }
  __syncthreads();

  const int wave = tid >> 5;
  const int lane = tid & 31;
  const int wt = blockIdx.x * (blockDim.x >> 5) + wave;
  if (wt >= n_tiles) return;
  const int p0 = wt * 16;
  float* ht = htiles + wave * (16 * HDIM);

  // stage h tile, also emit sequence slot 0 (prev_path_state)
#pragma unroll 1
  for (int r = 0; r < 16; ++r) {
    v2f d = *(const v2f*)(path_state + (size_t)(p0 + r) * HDIM + lane * 2);
    *(v2f*)(ht + r * HDIM + lane * 2) = d;
    *(v2f*)(seq + ((size_t)(p0 + r) * 9 + 0) * HDIM + lane * 2) = d;
  }

  const int half = (lane >> 4) & 1;
  const int row  = lane & 15;

#pragma unroll 1
  for (int t = 0; t < PLEN; ++t) {
    const int link = l2p[(size_t)(p0 + row) * PLEN + t];
    const float* xb = link_state + (size_t)link * HDIM + half * 2;
    gru_step_wave(xb, ht, KT, RKT, b0, b1, lane);
#pragma unroll 1
    for (int r = 0; r < 16; ++r) {
      v2f d = *(const v2f*)(ht + r * HDIM + lane * 2);
      *(v2f*)(seq + ((size_t)(p0 + r) * 9 + (t + 1)) * HDIM + lane * 2) = d;
    }
  }
#pragma unroll 1
  for (int r = 0; r < 16; ++r) {
    v2f d = *(const v2f*)(ht + r * HDIM + lane * 2);
    *(v2f*)(path_state + (size_t)(p0 + r) * HDIM + lane * 2) = d;
  }
}

// ---------------------------------------------------------------------------
// Link GRU: one wave = 16 links, single GRU step (x = path_sum).
// ---------------------------------------------------------------------------
__global__ void __launch_bounds__(512, 1)
link_gru_kernel(const float* __restrict__ path_sum,
                float* __restrict__ link_state,
                const float* __restrict__ luK,
                const float* __restrict__ luRK,
                const float* __restrict__ lub,
                int n_tiles)
{
  extern __shared__ float smem[];
  float* KT  = smem;
  float* RKT = KT + HDIM * G3;
  float* b0  = RKT + HDIM * G3;
  float* b1  = b0 + G3;
  float* htiles = b1 + G3;

  const int tid = threadIdx.x;
  for (int i = tid; i < HDIM * G3; i += blockDim.x) {
    int r = i / G3, c = i % G3;
    KT[c * HDIM + r]  = luK[i];
    RKT[c * HDIM + r] = luRK[i];
  }
  for (int i = tid; i < G3; i += blockDim.x) { b0[i] = lub[i]; b1[i] = lub[G3 + i]; }
  __syncthreads();

  const int wave = tid >> 5;
  const int lane = tid & 31;
  const int wt = blockIdx.x * (blockDim.x >> 5) + wave;
  if (wt >= n_tiles) return;
  const int l0 = wt * 16;
  float* ht = htiles + wave * (16 * HDIM);

#pragma unroll 1
  for (int r = 0; r < 16; ++r) {
    v2f d = *(const v2f*)(link_state + (size_t)(l0 + r) * HDIM + lane * 2);
    *(v2f*)(ht + r * HDIM + lane * 2) = d;
  }
  const int half = (lane >> 4) & 1;
  const float* xb = path_sum + (size_t)(l0 + (lane & 15)) * HDIM + half * 2;
  gru_step_wave(xb, ht, KT, RKT, b0, b1, lane);
#pragma unroll 1
  for (int r = 0; r < 16; ++r) {
    v2f d = *(const v2f*)(ht + r * HDIM + lane * 2);
    *(v2f*)(link_state + (size_t)(l0 + r) * HDIM + lane * 2) = d;
  }
}

// ---------------------------------------------------------------------------
// path_sum[l] = sum over DEG of seq[p2l[l,d,0], p2l[l,d,1], :]
// ---------------------------------------------------------------------------
__global__ void path_sum_kernel(const float* __restrict__ seq,
                                const int*   __restrict__ p2l,
                                float* __restrict__ path_sum, int n_links)
{
  const int col  = threadIdx.x & (HDIM - 1);
  const int sub  = threadIdx.x >> 6;          // 0..3
  const int link = blockIdx.x * 4 + sub;
  if (link >= n_links) return;
  float acc = 0.0f;
  for (int d = 0; d < DEG; ++d) {
    const int p   = p2l[((size_t)link * DEG + d) * 2 + 0];
    const int pos = p2l[((size_t)link * DEG + d) * 2 + 1];
    acc += seq[((size_t)p * 9 + pos) * HDIM + col];
  }
  path_sum[(size_t)link * HDIM + col] = acc;
}

// ---------------------------------------------------------------------------
// Link embedding: load = sum(traffic)/(cap*1e9); 2->64 gelu -> 64->64 gelu.
// One 64-thread block per link.
// ---------------------------------------------------------------------------
__global__ void link_embed_kernel(const float* __restrict__ flow_traffic,
                                  const float* __restrict__ link_capacity,
                                  const int*   __restrict__ p2l,
                                  const float* __restrict__ W1, const float* __restrict__ B1,
                                  const float* __restrict__ W2, const float* __restrict__ B2,
                                  float* __restrict__ link_state)
{
  __shared__ float red[HDIM];
  __shared__ float h1s[HDIM];
  const int link = blockIdx.x;
  const int j = threadIdx.x;
  float part = 0.0f;
  for (int d = j; d < DEG; d += HDIM)
    part += flow_traffic[p2l[((size_t)link * DEG + d) * 2 + 0]];
  red[j] = part;
  __syncthreads();
  for (int s = HDIM / 2; s > 0; s >>= 1) {
    if (j < s) red[j] += red[j + s];
    __syncthreads();
  }
  const float cap  = link_capacity[link];
  const float load = red[0] / (cap * 1.0e9f);
  h1s[j] = gelu_f(cap * W1[j] + load * W1[HDIM + j] + B1[j]);
  __syncthreads();
  float o = B2[j];
  for (int k = 0; k < HDIM; ++k) o = fmaf(h1s[k], W2[k * HDIM + j], o);
  link_state[(size_t)link * HDIM + j] = gelu_f(o);
}

// ---------------------------------------------------------------------------
// Path embedding: 5 features -> 64 gelu -> 64 gelu. 4 paths per block.
// ---------------------------------------------------------------------------
__global__ void path_embed_kernel(const float* __restrict__ ft,  const float* __restrict__ fp,
                                  const float* __restrict__ fps, const float* __restrict__ ipm,
                                  const float* __restrict__ ipv,
                                  const float* __restrict__ W1, const float* __restrict__ B1,
                                  const float* __restrict__ W2, const float* __restrict__ B2,
                                  float* __restrict__ path_state, int n_paths)
{
  __shared__ float h1s[4][HDIM];
  const int j = threadIdx.x;
  const int s = threadIdx.y;
  const int p = blockIdx.x * 4 + s;
  if (p < n_paths) {
    const float f0 = ft[p], f1 = fp[p], f2 = fps[p], f3 = ipm[p], f4 = ipv[p];
    float h = B1[j];
    h = fmaf(f0, W1[0 * HDIM + j], h);
    h = fmaf(f1, W1[1 * HDIM + j], h);
    h = fmaf(f2, W1[2 * HDIM + j], h);
    h = fmaf(f3, W1[3 * HDIM + j], h);
    h = fmaf(f4, W1[4 * HDIM + j], h);
    h1s[s][j] = gelu_f(h);
  }
  __syncthreads();
  if (p < n_paths) {
    float o = B2[j];
    for (int k = 0; k < HDIM; ++k) o = fmaf(h1s[s][k], W2[k * HDIM + j], o);
    path_state[(size_t)p * HDIM + j] = gelu_f(o);
  }
}

// ---------------------------------------------------------------------------
// Readout: 64->32 gelu ->16 gelu ->1 softplus; delay = sum_t occ/cap.
// One thread per path.
// ---------------------------------------------------------------------------
#define R1 32
#define R2 16
__global__ void readout_kernel(const float* __restrict__ seq,
                               const int*   __restrict__ l2p,
                               const float* __restrict__ link_capacity,
                               const float* __restrict__ W1, const float* __restrict__ B1,
                               const float* __restrict__ W2, const float* __restrict__ B2,
                               const float* __restrict__ W3, const float* __restrict__ B3,
                               float* __restrict__ delay, int n_paths)
{
  __shared__ float w1s[HDIM * R1];
  __shared__ float w2s[R1 * R2];
  __shared__ float w3s[R2];
  __shared__ float b1s[R1];
  __shared__ float b2s[R2];
  __shared__ float b3s;
  const int tid = threadIdx.x;
  for (int i = tid; i < HDIM * R1; i += blockDim.x) w1s[i] = W1[i];
  for (int i = tid; i < R1 * R2;  i += blockDim.x) w2s[i] = W2[i];
  if (tid < R2) { w3s[tid] = W3[tid]; b2s[tid] = B2[tid]; }
  if (tid < R1) b1s[tid] = B1[tid];
  if (tid == 0) b3s = B3[0];
  __syncthreads();
  const int p = blockIdx.x * blockDim.x + tid;
  if (p >= n_paths) return;
  float dsum = 0.0f;
  for (int t = 1; t <= PLEN; ++t) {
    const float* x = seq + ((size_t)p * 9 + t) * HDIM;
    float h1[R1];
#pragma unroll
    for (int jj = 0; jj < R1; ++jj) h1[jj] = b1s[jj];
    for (int k = 0; k < HDIM; ++k) {
      const float xv = x[k];
#pragma unroll
      for (int jj = 0; jj < R1; ++jj) h1[jj] = fmaf(xv, w1s[k * R1 + jj], h1[jj]);
    }
#pragma unroll
    for (int jj = 0; jj < R1; ++jj) h1[jj] = gelu_f(h1[jj]);
    float h2[R2];
#pragma unroll
    for (int jj = 0; jj < R2; ++jj) h2[jj] = b2s[jj];
    for (int k = 0; k < R1; ++k) {
#pragma unroll
      for (int jj = 0; jj < R2; ++jj) h2[jj] = fmaf(h1[k], w2s[k * R2 + jj], h2[jj]);
    }
    float o = b3s;
#pragma unroll
    for (int k = 0; k < R2; ++k) o = fmaf(gelu_f(h2[k]), w3s[k], o);
    const float occ  = softplus_f(o);
    const int  link  = l2p[(size_t)p * PLEN + (t - 1)];
    dsum += occ / link_capacity[link];
  }
  delay[p] = dsum;
}

// ---------------------------------------------------------------------------
extern "C" void kernel_launch(void* const* d_in, const int* in_sizes, int n_in,
                              void* d_out, int out_size, void* d_ws, size_t ws_size,
                              hipStream_t stream)
{
  const float* flow_traffic     = (const float*)d_in[0];
  const float* flow_packets     = (const float*)d_in[1];
  const float* flow_packet_size = (const float*)d_in[2];
  const float* link_capacity    = (const float*)d_in[3];
  const float* ipg_mean         = (const float*)d_in[4];
  const float* ipg_var          = (const float*)d_in[5];
  const int*   l2p              = (const int*)d_in[6];
  const int*   p2l              = (const int*)d_in[7];
  const float* peW1 = (const float*)d_in[8];
  const float* peb1 = (const float*)d_in[9];
  const float* peW2 = (const float*)d_in[10];
  const float* peb2 = (const float*)d_in[11];
  const float* leW1 = (const float*)d_in[12];
  const float* leb1 = (const float*)d_in[13];
  const float* leW2 = (const float*)d_in[14];
  const float* leb2 = (const float*)d_in[15];
  const float* puK  = (const float*)d_in[16];
  const float* puRK = (const float*)d_in[17];
  const float* pub  = (const float*)d_in[18];
  const float* luK  = (const float*)d_in[19];
  const float* luRK = (const float*)d_in[20];
  const float* lub  = (const float*)d_in[21];
  const float* roW1 = (const float*)d_in[22];
  const float* rob1 = (const float*)d_in[23];
  const float* roW2 = (const float*)d_in[24];
  const float* rob2 = (const float*)d_in[25];
  const float* roW3 = (const float*)d_in[26];
  const float* rob3 = (const float*)d_in[27];

  const int n_paths = in_sizes[0];
  const int n_links = in_sizes[3];

  // Workspace layout (f32): seq[n_paths*9*64] | path_state | link_state | path_sum
  float* seq        = (float*)d_ws;
  float* path_state = seq + (size_t)n_paths * 9 * HDIM;
  float* link_state = path_state + (size_t)n_paths * HDIM;
  float* path_sum   = link_state + (size_t)n_links * HDIM;
  float* delay      = (float*)d_out;

  link_embed_kernel<<<n_links, HDIM, 0, stream>>>(
      flow_traffic, link_capacity, p2l, leW1, leb1, leW2, leb2, link_state);

  path_embed_kernel<<<(n_paths + 3) / 4, dim3(HDIM, 4), 0, stream>>>(
      flow_traffic, flow_packets, flow_packet_size, ipg_mean, ipg_var,
      peW1, peb1, peW2, peb2, path_state, n_paths);

  const int ptiles = (n_paths + 15) / 16;
  const int ltiles = (n_links + 15) / 16;
  const size_t smem = (size_t)SMEM_FLOATS * sizeof(float);

  for (int it = 0; it < GITERS; ++it) {
    path_rnn_kernel<<<(ptiles + 15) / 16, 512, smem, stream>>>(
        link_state, l2p, path_state, seq, puK, puRK, pub, ptiles);
    if (it + 1 < GITERS) {   // final link update is unused by the output
      path_sum_kernel<<<(n_links + 3) / 4, 256, 0, stream>>>(seq, p2l, path_sum, n_links);
      link_gru_kernel<<<(ltiles + 15) / 16, 512, smem, stream>>>(
          path_sum, link_state, luK, luRK, lub, ltiles);
    }
  }

  readout_kernel<<<(n_paths + 255) / 256, 256, 0, stream>>>(
      seq, l2p, link_capacity, roW1, rob1, roW2, rob2, roW3, rob3, delay, n_paths);
}